// TwoLayerRGCN_14181982011501
// MI455X (gfx1250) — compile-verified
//
#include <hip/hip_runtime.h>
#include <hip/hip_bf16.h>
#include <stdint.h>

// ---------------------------------------------------------------------------
// Two-layer RGCN for MI455X (gfx1250, wave32, WMMA).
//
//   agg_i = sum_{e:dst=i} (x[src(e)] @ W[rel(e)]) * (1/cnt[rel(e), i])
//
// Edges binned by relation (device counting sort, 64-padded per relation);
// each wave owns four 16-edge WMMA tiles of one relation, loading the
// relation's B fragments (W_r) into registers once. All gather loads are
// unconditional with clamped indices — padding rows produce garbage that is
// never scattered (dst < 0 guard), avoiding exec-mask branches around loads.
// f32 V_WMMA_F32_16X16X4_F32 throughout: problem is HBM/atomic bound
// (~0.5 GB, ~6 GFLOP), so fp32 matrix ops are the right precision at 23.3TB/s.
// __launch_bounds__(128, 1) relaxes the occupancy-driven VGPR budget so the
// ~140-VGPR working set (B frags resident across 4 tiles) doesn't spill.
// ---------------------------------------------------------------------------

typedef float v2f __attribute__((ext_vector_type(2)));
typedef float v8f __attribute__((ext_vector_type(8)));

#define IN_C  64
#define HID_C 32
#define OUT_C 16

#define SEG_PAD 64            // per-relation segment padding (4 WMMA tiles)
#define TILES_PER_GROUP 4
#define WPB 4                 // waves per block (128 threads)

// --------------------------------- utility ---------------------------------

__global__ void fill_u32_kernel(unsigned* __restrict__ p, unsigned val, size_t n) {
    size_t i = (size_t)blockIdx.x * blockDim.x + threadIdx.x;
    if (i < n) p[i] = val;
}

__global__ void count_edges_kernel(const int* __restrict__ ei,   // [2,E] flat
                                   const int* __restrict__ et,   // [E]
                                   int* __restrict__ cnt,        // [R*N]
                                   int* __restrict__ relCount,   // [R]
                                   int N, int E) {
    int e = blockIdx.x * blockDim.x + threadIdx.x;
    if (e >= E) return;
    int dst = ei[E + e];
    int r   = et[e];
    __hip_atomic_fetch_add(&cnt[(size_t)r * N + dst], 1,
                           __ATOMIC_RELAXED, __HIP_MEMORY_SCOPE_AGENT);
    __hip_atomic_fetch_add(&relCount[r], 1,
                           __ATOMIC_RELAXED, __HIP_MEMORY_SCOPE_AGENT);
}

// Tiny serial exclusive scan; each relation's segment rounded up to SEG_PAD.
__global__ void scan_rel_kernel(const int* __restrict__ relCount,
                                int* __restrict__ relBase, int R) {
    if (threadIdx.x == 0 && blockIdx.x == 0) {
        int base = 0;
        for (int r = 0; r < R; ++r) {
            relBase[r] = base;
            base += (relCount[r] + (SEG_PAD - 1)) & ~(SEG_PAD - 1);
        }
        relBase[R] = base;
    }
}

// In-place: int count -> float reciprocal of max(count,1).
__global__ void count_to_inv_kernel(float* __restrict__ cntf, size_t n) {
    size_t i = (size_t)blockIdx.x * blockDim.x + threadIdx.x;
    if (i < n) {
        int c = ((const int*)cntf)[i];
        cntf[i] = 1.0f / (float)(c > 0 ? c : 1);
    }
}

__global__ void scatter_edges_kernel(const int* __restrict__ ei,
                                     const int* __restrict__ et,
                                     const int* __restrict__ relBase,
                                     int* __restrict__ cursor,
                                     int* __restrict__ srcS,
                                     int* __restrict__ dstS, int E) {
    int e = blockIdx.x * blockDim.x + threadIdx.x;
    if (e >= E) return;
    int src = ei[e];
    int dst = ei[E + e];
    int r   = et[e];
    int pos = relBase[r] + __hip_atomic_fetch_add(&cursor[r], 1,
                   __ATOMIC_RELAXED, __HIP_MEMORY_SCOPE_AGENT);
    srcS[pos] = src;
    dstS[pos] = dst;
}

// ------------------------- WMMA edge-message kernel -------------------------
template <int INC, int OUTC>
__global__ __launch_bounds__(WPB * 32, 1)
void rgcn_msg_wmma(const float* __restrict__ x,      // [N, INC]
                   const float* __restrict__ W,      // [R, INC, OUTC]
                   const int*   __restrict__ srcS,
                   const int*   __restrict__ dstS,
                   const int*   __restrict__ relBase,// [R+1]
                   const float* __restrict__ cntInv, // [R*N]
                   float*       __restrict__ agg,    // [N, OUTC]
                   int N, int R, int numGroups) {
    const int lane = threadIdx.x & 31;
    const int wave = threadIdx.x >> 5;
    const int grp  = blockIdx.x * (blockDim.x >> 5) + wave;
    if (grp >= numGroups) return;
    const int grpBase = grp * (16 * TILES_PER_GROUP);

    // Binary search: largest r with relBase[r] <= grpBase.
    int lo = 0, hi = R;
    while (lo + 1 < hi) {
        int mid = (lo + hi) >> 1;
        if (relBase[mid] <= grpBase) lo = mid; else hi = mid;
    }
    const int rel = lo;

    const int m  = lane & 15;            // row (edge within tile) / column id
    const int hh = (lane >> 4) & 1;      // lane half selects K sub-pair

    constexpr int KS = INC / 4;          // K-steps
    constexpr int NT = OUTC / 16;        // N tiles

    // B fragments: loaded ONCE per 64-edge group, reused by 4 sub-tiles.
    // f32 B 4x16 layout (mirrors A): VGPR0 = rows {0|2}, VGPR1 = rows {1|3}.
    v2f bfrag[KS][NT];
    const float* Wr = W + (size_t)rel * INC * OUTC;
#pragma unroll
    for (int k = 0; k < KS; ++k) {
#pragma unroll
        for (int t = 0; t < NT; ++t) {
            const int n = t * 16 + m;
            v2f b;
            b.x = Wr[(4 * k + 2 * hh)     * OUTC + n];
            b.y = Wr[(4 * k + 2 * hh + 1) * OUTC + n];
            bfrag[k][t] = b;
        }
    }

#pragma unroll 1   // keep one tile live at a time: no spills, compact code
    for (int s = 0; s < TILES_PER_GROUP; ++s) {
        const int tileBase = grpBase + s * 16;

        // Unconditional loads with clamped indices. Padding rows (src == -1)
        // produce garbage results that are never scattered (dst < 0 guard).
        const int srcRaw = srcS[tileBase + m];
        const int dstRaw = dstS[tileBase + m];
        const int dstm   = (srcRaw >= 0) ? dstRaw : -1;     // v_cndmask
        const int srcCl  = (srcRaw >= 0) ? srcRaw : 0;
        const int dstCl  = (dstm   >= 0) ? dstm   : 0;
        const float invm = cntInv[(size_t)rel * N + dstCl];

        // A fragments. ISA f32 A 16x4 layout: VGPR0 = K{0|2}, VGPR1 = K{1|3}.
        v2f afrag[KS];
        const float* xr = x + (size_t)srcCl * INC;
#pragma unroll
        for (int k = 0; k < KS; ++k)
            afrag[k] = *(const v2f*)(xr + 4 * k + 2 * hh);

        v8f acc[NT];
#pragma unroll
        for (int t = 0; t < NT; ++t) {
            v8f z = {0.f, 0.f, 0.f, 0.f, 0.f, 0.f, 0.f, 0.f};
            acc[t] = z;
        }
#pragma unroll
        for (int k = 0; k < KS; ++k) {
#pragma unroll
            for (int t = 0; t < NT; ++t) {
                acc[t] = __builtin_amdgcn_wmma_f32_16x16x4_f32(
                    false, afrag[k], false, bfrag[k][t], (short)0, acc[t],
                    false, false);
            }
        }

        // C/D layout: lane covers column n = m, rows v + 8*hh (v = 0..7).
        float invRow[8];
        int   dstRow[8];
#pragma unroll
        for (int v = 0; v < 8; ++v) {
            const int srcLane = v + 8 * hh;
            invRow[v] = __shfl(invm, srcLane, 32);
            dstRow[v] = __shfl(dstm, srcLane, 32);
        }
#pragma unroll
        for (int t = 0; t < NT; ++t) {
            const int n = t * 16 + m;
#pragma unroll
            for (int v = 0; v < 8; ++v) {
                if (dstRow[v] >= 0) {
                    __hip_atomic_fetch_add(&agg[(size_t)dstRow[v] * OUTC + n],
                                           acc[t][v] * invRow[v],
                                           __ATOMIC_RELAXED,
                                           __HIP_MEMORY_SCOPE_AGENT);
                }
            }
        }
    }
}

// ----------------------- WMMA node (root) transform -------------------------
template <int INC, int OUTC, bool RELU>
__global__ __launch_bounds__(WPB * 32, 1)
void rgcn_node_wmma(const float* __restrict__ x,    // [N, INC]
                    const float* __restrict__ root, // [INC, OUTC]
                    const float* __restrict__ bias, // [OUTC]
                    const float* __restrict__ agg,  // [N, OUTC]
                    float*       __restrict__ out,  // [N, OUTC]
                    int N, int numGroups) {
    const int lane = threadIdx.x & 31;
    const int wave = threadIdx.x >> 5;
    const int grp  = blockIdx.x * (blockDim.x >> 5) + wave;
    if (grp >= numGroups) return;

    const int m  = lane & 15;
    const int hh = (lane >> 4) & 1;

    constexpr int KS = INC / 4;
    constexpr int NT = OUTC / 16;

    v2f bfrag[KS][NT];
#pragma unroll
    for (int k = 0; k < KS; ++k) {
#pragma unroll
        for (int t = 0; t < NT; ++t) {
            const int n = t * 16 + m;
            v2f b;
            b.x = root[(4 * k + 2 * hh)     * OUTC + n];
            b.y = root[(4 * k + 2 * hh + 1) * OUTC + n];
            bfrag[k][t] = b;
        }
    }

    float bn[NT];
#pragma unroll
    for (int t = 0; t < NT; ++t) bn[t] = bias[t * 16 + m];

#pragma unroll 1
    for (int s = 0; s < TILES_PER_GROUP; ++s) {
        const int tile = grp * TILES_PER_GROUP + s;
        if (tile * 16 >= N) break;   // wave-uniform

        // Clamped unconditional gather; tail rows (node >= N) never stored.
        const int node_m = tile * 16 + m;
        const int nodeCl = (node_m < N) ? node_m : (N - 1);

        v2f afrag[KS];
        const float* xr = x + (size_t)nodeCl * INC;
#pragma unroll
        for (int k = 0; k < KS; ++k)
            afrag[k] = *(const v2f*)(xr + 4 * k + 2 * hh);

        v8f acc[NT];
#pragma unroll
        for (int t = 0; t < NT; ++t) {
            v8f z = {0.f, 0.f, 0.f, 0.f, 0.f, 0.f, 0.f, 0.f};
            acc[t] = z;
        }
#pragma unroll
        for (int k = 0; k < KS; ++k) {
#pragma unroll
            for (int t = 0; t < NT; ++t) {
                acc[t] = __builtin_amdgcn_wmma_f32_16x16x4_f32(
                    false, afrag[k], false, bfrag[k][t], (short)0, acc[t],
                    false, false);
            }
        }

#pragma unroll
        for (int t = 0; t < NT; ++t) {
            const int n = t * 16 + m;
#pragma unroll
            for (int v = 0; v < 8; ++v) {
                const int node = tile * 16 + v + 8 * hh;
                if (node < N) {
                    float val = acc[t][v] + bn[t] + agg[(size_t)node * OUTC + n];
                    if (RELU) val = fmaxf(val, 0.0f);
                    out[(size_t)node * OUTC + n] = val;
                }
            }
        }
    }
}

// --------------------------------- launch ----------------------------------

extern "C" void kernel_launch(void* const* d_in, const int* in_sizes, int n_in,
                              void* d_out, int out_size, void* d_ws, size_t ws_size,
                              hipStream_t stream) {
    const float* x     = (const float*)d_in[0];
    const int*   ei    = (const int*)  d_in[1];  // [2,E] flat (row0=src, row1=dst)
    const int*   et    = (const int*)  d_in[2];  // [E]
    const float* W1    = (const float*)d_in[3];  // [R, IN_C, HID_C]
    const float* root1 = (const float*)d_in[4];
    const float* b1    = (const float*)d_in[5];
    const float* W2    = (const float*)d_in[6];  // [R, HID_C, OUT_C]
    const float* root2 = (const float*)d_in[7];
    const float* b2    = (const float*)d_in[8];
    float*       out   = (float*)d_out;
    (void)n_in; (void)out_size;

    const int N = in_sizes[0] / IN_C;
    const int E = in_sizes[2];
    const int R = in_sizes[3] / (IN_C * HID_C);

    // ---- workspace carve-up (~30 MB total) ----
    char* p = (char*)d_ws;
    auto carve = [&](size_t bytes) {
        char* q = p;
        p += (bytes + 255) & ~(size_t)255;
        return (void*)q;
    };
    int*   cnt      = (int*)carve((size_t)R * N * 4);  // reused as cntInv (float)
    float* cntInv   = (float*)cnt;
    int*   relCount = (int*)carve((size_t)R * 4);
    int*   relBase  = (int*)carve((size_t)(R + 1) * 4);
    int*   cursor   = (int*)carve((size_t)R * 4);
    const int Epad  = ((E + SEG_PAD - 1) & ~(SEG_PAD - 1)) + R * SEG_PAD;
    int*   srcS     = (int*)carve((size_t)Epad * 4);
    int*   dstS     = (int*)carve((size_t)Epad * 4);
    float* agg1     = (float*)carve((size_t)N * HID_C * 4);
    float* agg2     = (float*)carve((size_t)N * OUT_C * 4);
    float* h        = (float*)carve((size_t)N * HID_C * 4);
    (void)ws_size;

    auto fill = [&](void* ptr, unsigned v, size_t n) {
        fill_u32_kernel<<<(unsigned)((n + 255) / 256), 256, 0, stream>>>(
            (unsigned*)ptr, v, n);
    };

    // ---- graph preprocessing (shared by both layers) ----
    fill(cnt,      0u,          (size_t)R * N);
    fill(relCount, 0u,          (size_t)R);
    fill(cursor,   0u,          (size_t)R);
    fill(srcS,     0xFFFFFFFFu, (size_t)Epad);   // padding slots: src = -1
    fill(agg1,     0u,          (size_t)N * HID_C);
    fill(agg2,     0u,          (size_t)N * OUT_C);

    count_edges_kernel<<<(E + 255) / 256, 256, 0, stream>>>(ei, et, cnt, relCount, N, E);
    scan_rel_kernel<<<1, 32, 0, stream>>>(relCount, relBase, R);
    count_to_inv_kernel<<<(unsigned)(((size_t)R * N + 255) / 256), 256, 0, stream>>>(
        cntInv, (size_t)R * N);
    scatter_edges_kernel<<<(E + 255) / 256, 256, 0, stream>>>(
        ei, et, relBase, cursor, srcS, dstS, E);

    const int numGroups  = Epad / SEG_PAD;     // 64 edges per wave-group
    const int nodeTiles  = (N + 15) / 16;
    const int nodeGroups = (nodeTiles + TILES_PER_GROUP - 1) / TILES_PER_GROUP;

    // ---- layer 1: agg1 += msg; h = relu(x@root1 + b1 + agg1) ----
    rgcn_msg_wmma<IN_C, HID_C><<<(numGroups + WPB - 1) / WPB, WPB * 32, 0, stream>>>(
        x, W1, srcS, dstS, relBase, cntInv, agg1, N, R, numGroups);
    rgcn_node_wmma<IN_C, HID_C, true><<<(nodeGroups + WPB - 1) / WPB, WPB * 32, 0, stream>>>(
        x, root1, b1, agg1, h, N, nodeGroups);

    // ---- layer 2: agg2 += msg; out = h@root2 + b2 + agg2 ----
    rgcn_msg_wmma<HID_C, OUT_C><<<(numGroups + WPB - 1) / WPB, WPB * 32, 0, stream>>>(
        h, W2, srcS, dstS, relBase, cntInv, agg2, N, R, numGroups);
    rgcn_node_wmma<HID_C, OUT_C, false><<<(nodeGroups + WPB - 1) / WPB, WPB * 32, 0, stream>>>(
        h, root2, b2, agg2, out, N, nodeGroups);
}